// GraphConvolution_42391327212273
// MI455X (gfx1250) — compile-verified
//
#include <hip/hip_runtime.h>
#include <hip/hip_bf16.h>

// GCN layer for MI455X (gfx1250, wave32):
//   xw  = x @ W                    (fp32 WMMA 16x16x4, W staged in LDS via async copy)
//   out = relu(scatter_add(val[e] * xw[col[e]] -> row[e]))
//
// N_NODES = 50000 (= 3125 * 16, exact 16-row tiles)
// IN_DIM  = 128, OUT_DIM = 64, N_EDGES = 800000

typedef __attribute__((ext_vector_type(2))) float v2f;
typedef __attribute__((ext_vector_type(8))) float v8f;
typedef int v4i __attribute__((vector_size(16)));   // matches builtin's param type

#define IN_DIM  128
#define OUT_DIM 64

#if defined(__has_builtin)
#if __has_builtin(__builtin_amdgcn_global_load_async_to_lds_b128)
#define HAVE_ASYNC_LDS 1
#endif
#if __has_builtin(__builtin_amdgcn_s_wait_asynccnt)
#define HAVE_WAIT_ASYNC 1
#endif
#endif

// ---------------------------------------------------------------------------
// Kernel 1: dense GEMM xw = x @ W using V_WMMA_F32_16X16X4_F32.
// One wave (32 lanes) computes a 16x64 output strip: 4 accumulators of 16x16,
// K-loop of 32 steps (K=4 each). 256 threads = 8 waves per block.
// W (128x64 = 32 KB) is staged once per block into LDS with async copies so
// the steady-state B-operand reads come from the DS pipe, leaving VMEM for
// the streaming A-operand.
// ---------------------------------------------------------------------------
__global__ __launch_bounds__(256) void gcn_gemm_wmma(
    const float* __restrict__ x,     // [n_nodes, 128]
    const float* __restrict__ w,     // [128, 64]
    float* __restrict__ xw,          // [n_nodes, 64]
    int n_nodes)
{
    __shared__ float wlds[IN_DIM * OUT_DIM];         // 32 KB of the 320 KB WGP LDS

    // ---- stage W: 8192 floats = 2048 float4, 8 per thread -----------------
    {
        const int tid = threadIdx.x;
#if defined(HAVE_ASYNC_LDS)
        #pragma unroll
        for (int i = 0; i < 8; ++i) {
            const int idx4 = tid + i * 256;          // float4 index
            __builtin_amdgcn_global_load_async_to_lds_b128(
                (__attribute__((address_space(1))) v4i*)(w + idx4 * 4),
                (__attribute__((address_space(3))) v4i*)(&wlds[idx4 * 4]),
                0, 0);
        }
#if defined(HAVE_WAIT_ASYNC)
        __builtin_amdgcn_s_wait_asynccnt(0);
#else
        asm volatile("s_wait_asynccnt 0x0" ::: "memory");
#endif
#else
        #pragma unroll
        for (int i = 0; i < 8; ++i) {
            const int idx4 = tid + i * 256;
            ((float4*)wlds)[idx4] = ((const float4*)w)[idx4];
        }
#endif
        __syncthreads();
    }

    const int lane  = threadIdx.x & 31;
    const int wave  = threadIdx.x >> 5;              // 0..7
    const int mtile = blockIdx.x * 8 + wave;         // 16-row tile index
    const int m0    = mtile * 16;
    if (m0 >= n_nodes) return;                       // wave-uniform: EXEC stays all-1s

    // A-matrix 16x4 f32 layout: lanes 0-15 -> M=lane, K={0,1}; lanes 16-31 -> M=lane-16, K={2,3}
    const int mrow  = lane & 15;
    const int khalf = (lane >> 4) << 1;              // 0 or 2
    const long long arow = (long long)(m0 + mrow) * IN_DIM;

    v8f acc0 = {}, acc1 = {}, acc2 = {}, acc3 = {};

    #pragma unroll 4
    for (int k0 = 0; k0 < IN_DIM; k0 += 4) {
        v2f a;
        a.x = x[arow + k0 + khalf];
        a.y = x[arow + k0 + khalf + 1];

        // B-matrix 4x16 f32 layout (mirror of A): lanes 0-15 -> N=lane, K={0,1};
        // lanes 16-31 -> N=lane-16, K={2,3}.  Read from LDS-resident W.
        const int n = lane & 15;
        const int krow0 = (k0 + khalf) * OUT_DIM;
        const int krow1 = krow0 + OUT_DIM;

        v2f b0, b1, b2, b3;
        b0.x = wlds[krow0 + 0  + n];  b0.y = wlds[krow1 + 0  + n];
        b1.x = wlds[krow0 + 16 + n];  b1.y = wlds[krow1 + 16 + n];
        b2.x = wlds[krow0 + 32 + n];  b2.y = wlds[krow1 + 32 + n];
        b3.x = wlds[krow0 + 48 + n];  b3.y = wlds[krow1 + 48 + n];

        // 8 args: (neg_a, A, neg_b, B, c_mod, C, reuse_a, reuse_b)
        acc0 = __builtin_amdgcn_wmma_f32_16x16x4_f32(false, a, false, b0, (short)0, acc0, false, false);
        acc1 = __builtin_amdgcn_wmma_f32_16x16x4_f32(false, a, false, b1, (short)0, acc1, false, false);
        acc2 = __builtin_amdgcn_wmma_f32_16x16x4_f32(false, a, false, b2, (short)0, acc2, false, false);
        acc3 = __builtin_amdgcn_wmma_f32_16x16x4_f32(false, a, false, b3, (short)0, acc3, false, false);
    }

    // C/D 16x16 f32 layout: VGPR r -> M=r (lanes 0-15), M=r+8 (lanes 16-31); N=lane&15.
    const int ncol  = lane & 15;
    const int rbase = (lane >> 4) * 8;
    #pragma unroll
    for (int r = 0; r < 8; ++r) {
        const long long orow = (long long)(m0 + rbase + r) * OUT_DIM + ncol;
        xw[orow + 0 ] = acc0[r];
        xw[orow + 16] = acc1[r];
        xw[orow + 32] = acc2[r];
        xw[orow + 48] = acc3[r];
    }
}

// ---------------------------------------------------------------------------
// Kernel 0: zero the output accumulator (d_out is poisoned by the harness).
// ---------------------------------------------------------------------------
__global__ void gcn_zero(float4* __restrict__ out, long long n4)
{
    long long i = (long long)blockIdx.x * blockDim.x + threadIdx.x;
    if (i < n4) out[i] = make_float4(0.f, 0.f, 0.f, 0.f);
}

// ---------------------------------------------------------------------------
// Kernel 2: COO SpMM scatter.  16 lanes cover one edge (4 dims/thread):
// one global_load_b128 gather of xw[col], 4 consecutive-address fp32 atomics
// into out[row].  xw and out (12.8 MB each) are L2-resident (192 MB L2), so
// both the gather and the atomic RMW stay on-die.
// ---------------------------------------------------------------------------
__global__ __launch_bounds__(256) void gcn_scatter(
    const int*   __restrict__ adj_row,
    const int*   __restrict__ adj_col,
    const float* __restrict__ adj_val,
    const float* __restrict__ xw,
    float*       __restrict__ out,
    long long n_work)                                // n_edges * 16
{
    long long gid = (long long)blockIdx.x * blockDim.x + threadIdx.x;
    if (gid >= n_work) return;
    const int e = (int)(gid >> 4);
    const int d = (int)(gid & 15) * 4;               // dim base: 0,4,...,60

    const int   r = adj_row[e];
    const int   c = adj_col[e];
    const float v = adj_val[e];

    const float4 g = *(const float4*)&xw[(long long)c * OUT_DIM + d];
    float* o = &out[(long long)r * OUT_DIM + d];
    unsafeAtomicAdd(o + 0, v * g.x);                 // global_atomic_add_f32 (no return)
    unsafeAtomicAdd(o + 1, v * g.y);
    unsafeAtomicAdd(o + 2, v * g.z);
    unsafeAtomicAdd(o + 3, v * g.w);
}

// ---------------------------------------------------------------------------
// Kernel 3: in-place ReLU on the aggregated output (float4-vectorized).
// ---------------------------------------------------------------------------
__global__ void gcn_relu(float4* __restrict__ out, long long n4)
{
    long long i = (long long)blockIdx.x * blockDim.x + threadIdx.x;
    if (i < n4) {
        float4 v = out[i];
        v.x = v.x > 0.f ? v.x : 0.f;
        v.y = v.y > 0.f ? v.y : 0.f;
        v.z = v.z > 0.f ? v.z : 0.f;
        v.w = v.w > 0.f ? v.w : 0.f;
        out[i] = v;
    }
}

// ---------------------------------------------------------------------------
extern "C" void kernel_launch(void* const* d_in, const int* in_sizes, int n_in,
                              void* d_out, int out_size, void* d_ws, size_t ws_size,
                              hipStream_t stream)
{
    const float* x       = (const float*)d_in[0];
    const float* w       = (const float*)d_in[1];
    const int*   adj_row = (const int*)  d_in[2];
    const int*   adj_col = (const int*)  d_in[3];
    const float* adj_val = (const float*)d_in[4];

    float* out = (float*)d_out;
    float* xw  = (float*)d_ws;                       // needs n_nodes*64*4 = 12.8 MB

    const int n_nodes = in_sizes[0] / IN_DIM;        // 50000
    const int n_edges = in_sizes[2];                 // 800000

    // 0) zero output accumulator (float4-vectorized)
    {
        long long n4 = (long long)n_nodes * OUT_DIM / 4;
        int blk = 256;
        long long grd = (n4 + blk - 1) / blk;
        gcn_zero<<<(unsigned)grd, blk, 0, stream>>>((float4*)out, n4);
    }

    // 1) xw = x @ W   (WMMA f32 16x16x4, LDS-staged W)
    {
        int mtiles = (n_nodes + 15) / 16;            // 3125
        int blocks = (mtiles + 7) / 8;               // 8 waves/block
        gcn_gemm_wmma<<<blocks, 256, 0, stream>>>(x, w, xw, n_nodes);
    }

    // 2) scatter-add messages (16 lanes per edge, float4 gather)
    {
        long long n_work = (long long)n_edges * 16;
        int blk = 256;
        long long grd = (n_work + blk - 1) / blk;
        gcn_scatter<<<(unsigned)grd, blk, 0, stream>>>(adj_row, adj_col, adj_val,
                                                       xw, out, n_work);
    }

    // 3) ReLU in place
    {
        long long n4 = (long long)n_nodes * OUT_DIM / 4;
        int blk = 256;
        long long grd = (n4 + blk - 1) / blk;
        gcn_relu<<<(unsigned)grd, blk, 0, stream>>>((float4*)out, n4);
    }
}